// MambaExpert_33749853012617
// MI455X (gfx1250) — compile-verified
//
#include <hip/hip_runtime.h>
#include <hip/hip_bf16.h>
#include <math.h>

// ---------------- problem constants (from reference) ----------------
#define D_MODEL 1024
#define D_STATE 16
#define D_CONV  4
#define D_INNER 2048
#define DT_RANK 64
#define BATCH   2
#define SEQ     2048
#define MROWS   (BATCH * SEQ)            // 4096 rows for all GEMMs
#define XZ_N    (2 * D_INNER)            // 4096
#define DBC_N   (DT_RANK + 2 * D_STATE)  // 96

typedef __attribute__((ext_vector_type(2))) float v2f;
typedef __attribute__((ext_vector_type(8))) float v8f;

__device__ __forceinline__ float sigmoid_f(float x) { return 1.0f / (1.0f + __expf(-x)); }
__device__ __forceinline__ float silu_f(float x)    { return x * sigmoid_f(x); }
__device__ __forceinline__ float softplus_f(float x){ return (x > 20.0f) ? x : log1pf(__expf(x)); }

// ---------------------------------------------------------------
// Register-blocked fp32 WMMA GEMM:
//   C[M,N] = A[M,K] (row-major, stride lda)  x  W[N,K]^T
// Each wave owns an MB x NB grid of 16x16 C tiles, stepping K by 4
// with V_WMMA_F32_16X16X4_F32.  Per K-step: MB+NB b64 loads feed
// MB*NB WMMAs (0.75 loads/wmma at 2x4 vs 2.0 unblocked).
// EPI=1 fuses softplus(acc + bias[n]).
// Fragment mapping (ISA 7.12.2, 32-bit 16x4):
//   A: lane L holds row m=L%16, k = kk + 2*(L/16) + {0,1}
//   B: lane L holds col n=L%16, same k split
//   C/D: VGPR i -> (m = i + 8*(L/16), n = L%16)
// ---------------------------------------------------------------
template <int MB, int NB, int EPI>
__global__ void wmma_gemm_f32(const float* __restrict__ A, int lda,
                              const float* __restrict__ W,   // [N,K]
                              float* __restrict__ C, int ldc,
                              const float* __restrict__ bias,
                              int M, int N, int K)
{
    const int lane  = threadIdx.x & 31;
    const int wave  = threadIdx.x >> 5;
    const int Ng    = N / (16 * NB);
    const int Mg    = M / (16 * MB);
    const int group = blockIdx.x * (blockDim.x >> 5) + wave;
    if (group >= Mg * Ng) return;                 // wave-uniform, never taken

    const int tm   = (group / Ng) * (16 * MB);
    const int tn   = (group % Ng) * (16 * NB);
    const int half = lane >> 4;                   // 0 or 1
    const int l16  = lane & 15;

    const float* __restrict__ Arow[MB];
    const float* __restrict__ Wrow[NB];
#pragma unroll
    for (int mi = 0; mi < MB; ++mi)
        Arow[mi] = A + (size_t)(tm + mi * 16 + l16) * lda;
#pragma unroll
    for (int ni = 0; ni < NB; ++ni)
        Wrow[ni] = W + (size_t)(tn + ni * 16 + l16) * K;

    v8f acc[MB][NB];
#pragma unroll
    for (int mi = 0; mi < MB; ++mi)
#pragma unroll
        for (int ni = 0; ni < NB; ++ni)
            acc[mi][ni] = (v8f){0.f, 0.f, 0.f, 0.f, 0.f, 0.f, 0.f, 0.f};

    for (int kk = 0; kk < K; kk += 4) {
        const int k0 = kk + (half << 1);
        v2f a[MB], b[NB];
#pragma unroll
        for (int mi = 0; mi < MB; ++mi) { a[mi].x = Arow[mi][k0]; a[mi].y = Arow[mi][k0 + 1]; }
#pragma unroll
        for (int ni = 0; ni < NB; ++ni) { b[ni].x = Wrow[ni][k0]; b[ni].y = Wrow[ni][k0 + 1]; }
#pragma unroll
        for (int mi = 0; mi < MB; ++mi)
#pragma unroll
            for (int ni = 0; ni < NB; ++ni)
                acc[mi][ni] = __builtin_amdgcn_wmma_f32_16x16x4_f32(
                    /*neg_a=*/false, a[mi], /*neg_b=*/false, b[ni],
                    /*c_mod=*/(short)0, acc[mi][ni],
                    /*reuse_a=*/false, /*reuse_b=*/false);
    }

#pragma unroll
    for (int mi = 0; mi < MB; ++mi) {
        const int mbase = tm + mi * 16 + (half << 3);
#pragma unroll
        for (int ni = 0; ni < NB; ++ni) {
            const int n = tn + ni * 16 + l16;
            float bv = 0.0f;
            if (EPI == 1) bv = bias[n];
#pragma unroll
            for (int i = 0; i < 8; ++i) {
                float v = acc[mi][ni][i];
                if (EPI == 1) v = softplus_f(v + bv);
                C[(size_t)(mbase + i) * ldc + n] = v;
            }
        }
    }
}

// ---------------------------------------------------------------
// Depthwise causal conv (width 4) + bias + SiLU.
// x_in lives in xz[...,0:2048] (row stride 4096); u is dense [M,2048].
// ---------------------------------------------------------------
__global__ void conv_silu_kernel(const float* __restrict__ xz,
                                 const float* __restrict__ cw,   // [D_INNER,4]
                                 const float* __restrict__ cb,   // [D_INNER]
                                 float* __restrict__ u)
{
    const int idx = blockIdx.x * blockDim.x + threadIdx.x;   // over M*D_INNER
    if (idx >= MROWS * D_INNER) return;
    const int e  = idx & (D_INNER - 1);
    const int bt = idx >> 11;                                // /2048
    const int t  = bt & (SEQ - 1);
    const int b  = bt >> 11;

    float acc = cb[e];
#pragma unroll
    for (int j = 0; j < D_CONV; ++j) {
        const int tt = t - (D_CONV - 1) + j;
        if (tt >= 0)
            acc += xz[((size_t)(b * SEQ + tt)) * XZ_N + e] * cw[e * D_CONV + j];
    }
    u[idx] = silu_f(acc);
}

// ---------------------------------------------------------------
// Selective scan.  One lane per (b, e, n) recurrence: 65536 lanes.
// 16 state-lanes per channel; sum_n(h*C) via shfl_xor width-16.
// Fused epilogue: y = (scan + u*Dskip) * silu(z), written back into
// the (dead) x_in half of xz so out-proj reads it with lda=4096.
// ---------------------------------------------------------------
__global__ void scan_kernel(const float* __restrict__ dtp,   // [M, D_INNER]
                            const float* __restrict__ u,     // [M, D_INNER]
                            const float* __restrict__ xdbc,  // [M, 96]
                            const float* __restrict__ A_log, // [D_INNER, 16]
                            const float* __restrict__ Dskip, // [D_INNER]
                            float* xz)                       // read z / write y
{
    const int g    = blockIdx.x * blockDim.x + threadIdx.x;  // [0, 65536)
    const int n    = g & 15;
    const int chan = g >> 4;                                 // [0, 4096)
    const int e    = chan & (D_INNER - 1);
    const int b    = chan >> 11;

    const float A   = -__expf(A_log[e * D_STATE + n]);
    const float dsk = Dskip[e];
    float h = 0.0f;

    for (int t = 0; t < SEQ; ++t) {
        const size_t row = (size_t)(b * SEQ + t);
        const float dt_v = dtp[row * D_INNER + e];
        const float u_v  = u  [row * D_INNER + e];
        const float* dbc = xdbc + row * DBC_N;
        const float Bn = dbc[DT_RANK + n];
        const float Cn = dbc[DT_RANK + D_STATE + n];

        const float dA = __expf(dt_v * A);
        h = h * dA + dt_v * Bn * u_v;

        float part = h * Cn;
        part += __shfl_xor(part, 1, 16);
        part += __shfl_xor(part, 2, 16);
        part += __shfl_xor(part, 4, 16);
        part += __shfl_xor(part, 8, 16);

        if (n == 0) {
            const float z  = xz[row * XZ_N + D_INNER + e];
            xz[row * XZ_N + e] = (part + u_v * dsk) * silu_f(z);
        }
    }
}

// ---------------------------------------------------------------
extern "C" void kernel_launch(void* const* d_in, const int* in_sizes, int n_in,
                              void* d_out, int out_size, void* d_ws, size_t ws_size,
                              hipStream_t stream)
{
    const float* x      = (const float*)d_in[0];
    const float* W_in   = (const float*)d_in[1];
    const float* conv_w = (const float*)d_in[2];
    const float* conv_b = (const float*)d_in[3];
    const float* W_xproj= (const float*)d_in[4];
    const float* W_dt   = (const float*)d_in[5];
    const float* b_dt   = (const float*)d_in[6];
    const float* A_log  = (const float*)d_in[7];
    const float* D_skip = (const float*)d_in[8];
    const float* W_out  = (const float*)d_in[9];
    float* out = (float*)d_out;

    // workspace layout (floats): xz | u | x_dbc | dt   (~136 MB total)
    float* ws   = (float*)d_ws;
    float* xz   = ws;                                  // 4096 * 4096
    float* u    = xz   + (size_t)MROWS * XZ_N;         // 4096 * 2048
    float* xdbc = u    + (size_t)MROWS * D_INNER;      // 4096 * 96
    float* dt   = xdbc + (size_t)MROWS * DBC_N;        // 4096 * 2048

    // 1) in-projection: xz = x @ W_in^T        (4096 x 4096 x 1024), 32x64 blocks
    {
        const int groups = (MROWS / 32) * (XZ_N / 64);   // 128 * 64 = 8192
        wmma_gemm_f32<2, 4, 0><<<groups / 8, 256, 0, stream>>>(
            x, D_MODEL, W_in, xz, XZ_N, nullptr, MROWS, XZ_N, D_MODEL);
    }

    // 2) depthwise conv + SiLU -> u
    conv_silu_kernel<<<(MROWS * D_INNER) / 256, 256, 0, stream>>>(xz, conv_w, conv_b, u);

    // 3) x-projection: x_dbc = u @ W_xproj^T   (4096 x 96 x 2048), 16x96 blocks
    {
        const int groups = (MROWS / 16) * (DBC_N / 96);  // 256
        wmma_gemm_f32<1, 6, 0><<<groups / 8, 256, 0, stream>>>(
            u, D_INNER, W_xproj, xdbc, DBC_N, nullptr, MROWS, DBC_N, D_INNER);
    }

    // 4) dt-projection + softplus: dt = softplus(dt_lo @ W_dt^T + b_dt)
    //    dt_lo = x_dbc[:, 0:64]  (lda = 96, K = 64), 32x64 blocks
    {
        const int groups = (MROWS / 32) * (D_INNER / 64); // 128 * 32 = 4096
        wmma_gemm_f32<2, 4, 1><<<groups / 8, 256, 0, stream>>>(
            xdbc, DBC_N, W_dt, dt, D_INNER, b_dt, MROWS, D_INNER, DT_RANK);
    }

    // 5) selective scan + gate; writes y into xz's (dead) x_in half
    scan_kernel<<<(BATCH * D_INNER * 16) / 256, 256, 0, stream>>>(
        dt, u, xdbc, A_log, D_skip, xz);

    // 6) out-projection: out = y @ W_out^T     (y read from xz, lda = 4096), 32x64 blocks
    {
        const int groups = (MROWS / 32) * (D_MODEL / 64); // 128 * 16 = 2048
        wmma_gemm_f32<2, 4, 0><<<groups / 8, 256, 0, stream>>>(
            xz, XZ_N, W_out, out, D_MODEL, nullptr, MROWS, D_MODEL, D_INNER);
    }
}